// Predecessor_76081050682084
// MI455X (gfx1250) — compile-verified
//
#include <hip/hip_runtime.h>

#define HDIM 128
#define NEGV (-1000000000.0f)

typedef __attribute__((ext_vector_type(2))) float v2f;
typedef __attribute__((ext_vector_type(4))) float v4f;
typedef __attribute__((ext_vector_type(8))) float v8f;

// One wave (32 lanes) computes dd/ds for 16 rows of h via V_WMMA_F32_16X16X4_F32.
// B is a virtual 128x16 matrix: col0 = W[0:128] (dest weights), col1 = W[128:256]
// (source weights), other columns zero.
__global__ __launch_bounds__(32) void node_dots_wmma(const float* __restrict__ h,
                                                     const float* __restrict__ W,
                                                     float* __restrict__ dd,
                                                     float* __restrict__ ds) {
  const int lane  = threadIdx.x;        // 0..31, wave32
  const int row0  = blockIdx.x * 16;    // tile of 16 nodes
  const int m     = lane & 15;          // A-matrix row within tile (and B/C column)
  const int khalf = (lane >> 4) * 2;    // lanes 16..31 carry K+2,K+3 halves

  // Per-lane B source: column 0 -> W[0:128], column 1 -> W[128:256], else zero.
  const float* wp  = (m == 0) ? W : (W + HDIM);
  const float  msk = (m < 2) ? 1.0f : 0.0f;
  const float* arow = h + (size_t)(row0 + m) * HDIM;

  v8f c = {};   // 16x16 f32 accumulator tile (8 VGPRs)
  for (int k0 = 0; k0 < HDIM; k0 += 4) {
    const int kk = k0 + khalf;
    v2f a, bb;
    a.x  = arow[kk];
    a.y  = arow[kk + 1];
    bb.x = msk * wp[kk];
    bb.y = msk * wp[kk + 1];
    // (neg_a, A, neg_b, B, c_mod, C, reuse_a, reuse_b)
    c = __builtin_amdgcn_wmma_f32_16x16x4_f32(false, a, false, bb,
                                              (short)0, c, false, false);
  }

  // C layout: VGPR j holds M=j (lanes 0-15) / M=j+8 (lanes 16-31), N = lane&15.
  // Column N==0 is dd, column N==1 is ds.
  if (m < 2) {
    float* outv = (m == 0) ? dd : ds;
    const int mbase = row0 + ((lane >> 4) << 3);
#pragma unroll
    for (int j = 0; j < 8; ++j) outv[mbase + j] = c[j];
  }
}

// Streaming fill of the 604MB output with NEG using nontemporal b128 stores.
__global__ void fill_neg(float* __restrict__ out, long long n) {
  const long long n4 = n >> 2;
  long long i = (long long)blockIdx.x * blockDim.x + threadIdx.x;
  const long long stride = (long long)gridDim.x * blockDim.x;
  const v4f v = {NEGV, NEGV, NEGV, NEGV};
  v4f* __restrict__ out4 = (v4f*)out;
  for (; i < n4; i += stride) {
    __builtin_nontemporal_store(v, out4 + i);
  }
  // tail (out_size is 4-divisible here, but keep it safe)
  if (blockIdx.x == 0 && threadIdx.x == 0) {
    for (long long t = n4 << 2; t < n; ++t) out[t] = NEGV;
  }
}

// One thread per edge: out[d*N + s] = dd[d] + ds[s] + w_e * W[256] + b.
__global__ void scatter_edges(const int* __restrict__ src,
                              const int* __restrict__ dst,
                              const float* __restrict__ w,
                              const float* __restrict__ W,
                              const float* __restrict__ bias,
                              const float* __restrict__ dd,
                              const float* __restrict__ ds,
                              float* __restrict__ out,
                              int E, int Nn) {
  const int e = blockIdx.x * blockDim.x + threadIdx.x;
  if (e >= E) return;
  const int s = src[e];
  const int d = dst[e];
  const float val = dd[d] + ds[s] + w[e] * W[2 * HDIM] + bias[0];
  out[(size_t)d * (size_t)Nn + (size_t)s] = val;
}

extern "C" void kernel_launch(void* const* d_in, const int* in_sizes, int n_in,
                              void* d_out, int out_size, void* d_ws, size_t ws_size,
                              hipStream_t stream) {
  const float* h       = (const float*)d_in[0];  // [N, 128]
  const int*   sources = (const int*)d_in[1];    // [E]
  const int*   dests   = (const int*)d_in[2];    // [E]
  const float* weights = (const float*)d_in[3];  // [E, 1]
  const float* W       = (const float*)d_in[4];  // [1, 257]
  const float* b       = (const float*)d_in[5];  // [1]

  const int N = in_sizes[0] / HDIM;   // 12288
  const int E = in_sizes[1];          // 393216

  float* out = (float*)d_out;
  float* dd = (float*)d_ws;           // [N]
  float* ds = dd + N;                 // [N]

  // 1) per-node dot products via WMMA (tiny; 768 waves)
  node_dots_wmma<<<N / 16, 32, 0, stream>>>(h, W, dd, ds);

  // 2) streaming NEG fill of the full N*N output (bandwidth-bound pass)
  fill_neg<<<32768, 256, 0, stream>>>(out, (long long)out_size);

  // 3) sparse scatter of edge values (stream order serializes after the fill)
  scatter_edges<<<(E + 255) / 256, 256, 0, stream>>>(sources, dests, weights, W, b,
                                                     dd, ds, out, E, N);
}